// PTv3DownSample_26654567039606
// MI455X (gfx1250) — compile-verified
//
#include <hip/hip_runtime.h>
#include <hip/hip_bf16.h>
#include <math.h>

#define BB     8
#define NN     4096
#define KK     16
#define MM     2048
#define C_IN   64
#define C_OUT  128
#define EPSBN  1e-5f

typedef float v2f __attribute__((ext_vector_type(2)));
typedef float v8f __attribute__((ext_vector_type(8)));

// ---------------------------------------------------------------------------
// 1) Farthest point sampling: one block per cloud, pos cached in LDS.
// ---------------------------------------------------------------------------
__global__ __launch_bounds__(1024) void fps_kernel(const float* __restrict__ pos,
                                                   int* __restrict__ fps_idx) {
    const int b   = blockIdx.x;
    const int tid = threadIdx.x;

    __shared__ float spos[NN * 3];
    __shared__ float swval[32];
    __shared__ int   swidx[32];
    __shared__ int   s_last;

    const float* posb = pos + (size_t)b * NN * 3;
    for (int i = tid; i < NN * 3; i += 1024) spos[i] = posb[i];
    __syncthreads();

    // each thread owns 4 consecutive points
    float dmin[4];
    int   pts[4];
#pragma unroll
    for (int j = 0; j < 4; ++j) { pts[j] = tid * 4 + j; dmin[j] = 3.4e38f; }

    if (tid == 0) fps_idx[b * MM + 0] = 0;
    int last = 0;

    for (int step = 1; step < MM; ++step) {
        const float lx = spos[last * 3 + 0];
        const float ly = spos[last * 3 + 1];
        const float lz = spos[last * 3 + 2];

        float bestv = -1.0f;
        int   besti = 0x7fffffff;
#pragma unroll
        for (int j = 0; j < 4; ++j) {
            const int p = pts[j];
            const float dx = spos[p * 3 + 0] - lx;
            const float dy = spos[p * 3 + 1] - ly;
            const float dz = spos[p * 3 + 2] - lz;
            const float d = dx * dx + dy * dy + dz * dz;
            if (d < dmin[j]) dmin[j] = d;
            if (dmin[j] > bestv) { bestv = dmin[j]; besti = p; }   // ascending j keeps lowest idx on tie
        }
        // wave32 argmax reduction (lower lane = lower index wins ties)
#pragma unroll
        for (int off = 16; off > 0; off >>= 1) {
            const float ov = __shfl_down(bestv, off, 32);
            const int   oi = __shfl_down(besti, off, 32);
            if (ov > bestv || (ov == bestv && oi < besti)) { bestv = ov; besti = oi; }
        }
        const int wave = tid >> 5;
        if ((tid & 31) == 0) { swval[wave] = bestv; swidx[wave] = besti; }
        __syncthreads();
        if (wave == 0) {
            bestv = swval[tid & 31];
            besti = swidx[tid & 31];
#pragma unroll
            for (int off = 16; off > 0; off >>= 1) {
                const float ov = __shfl_down(bestv, off, 32);
                const int   oi = __shfl_down(besti, off, 32);
                if (ov > bestv || (ov == bestv && oi < besti)) { bestv = ov; besti = oi; }
            }
            if (tid == 0) { s_last = besti; fps_idx[b * MM + step] = besti; }
        }
        __syncthreads();
        last = s_last;
    }
}

// ---------------------------------------------------------------------------
// 2) KNN: block = 256 queries of one cloud, cloud pos cached in LDS.
//    Insertion-sorted top-16 (strict <) matches lax.top_k tie semantics.
// ---------------------------------------------------------------------------
__global__ __launch_bounds__(256) void knn_kernel(const float* __restrict__ pos,
                                                  const int* __restrict__ fps_idx,
                                                  int* __restrict__ nbr) {
    const int b = blockIdx.y;
    __shared__ float spos[NN * 3];
    const float* posb = pos + (size_t)b * NN * 3;
    for (int i = threadIdx.x; i < NN * 3; i += 256) spos[i] = posb[i];
    __syncthreads();

    const int q  = blockIdx.x * 256 + threadIdx.x;     // 0..MM-1
    const int qi = fps_idx[b * MM + q];
    const float qx = spos[qi * 3 + 0];
    const float qy = spos[qi * 3 + 1];
    const float qz = spos[qi * 3 + 2];

    float kd[KK];
    int   ki[KK];
#pragma unroll
    for (int j = 0; j < KK; ++j) { kd[j] = 3.4e38f; ki[j] = -1; }

    for (int p = 0; p < NN; ++p) {
        const float dx = spos[p * 3 + 0] - qx;
        const float dy = spos[p * 3 + 1] - qy;
        const float dz = spos[p * 3 + 2] - qz;
        const float d = dx * dx + dy * dy + dz * dz;
        if (d < kd[KK - 1]) {
            int j = KK - 1;
            while (j > 0 && kd[j - 1] > d) { kd[j] = kd[j - 1]; ki[j] = ki[j - 1]; --j; }
            kd[j] = d; ki[j] = p;
        }
    }
    int* out = nbr + ((size_t)(b * MM + q)) * KK;
#pragma unroll
    for (int j = 0; j < KK; ++j) out[j] = ki[j];
}

// ---------------------------------------------------------------------------
// 3) GEMM h = x @ W + b via V_WMMA_F32_16X16X4_F32. One wave per 16x16 tile.
// ---------------------------------------------------------------------------
__global__ __launch_bounds__(32) void gemm_kernel(const float* __restrict__ x,
                                                  const float* __restrict__ W,
                                                  const float* __restrict__ bias,
                                                  float* __restrict__ h) {
    const int tile = blockIdx.x;                // (B*N/16) * (C_OUT/16) tiles
    const int tm   = tile / (C_OUT / 16);
    const int tn   = tile % (C_OUT / 16);
    const int lane = threadIdx.x;
    const int lm   = lane & 15;
    const int hi   = lane >> 4;
    const int row0 = tm * 16;
    const int col0 = tn * 16;

    v8f acc = {};
    // A lane base: row = row0+lm, K base offset 2*hi (lanes16-31 hold K+2,K+3)
    const float* xrow = x + (size_t)(row0 + lm) * C_IN + 2 * hi;
    // B lane base: col = col0+lm, K base offset 2*hi
    const float* wcol = W + (size_t)(2 * hi) * C_OUT + col0 + lm;

#pragma unroll
    for (int k0 = 0; k0 < C_IN; k0 += 4) {
        v2f a = *(const v2f*)(xrow + k0);                    // K = k0+2*hi, k0+2*hi+1
        v2f bmat;
        bmat.x = wcol[(size_t)k0 * C_OUT];                   // K = k0+2*hi
        bmat.y = wcol[(size_t)(k0 + 1) * C_OUT];             // K = k0+2*hi+1
        acc = __builtin_amdgcn_wmma_f32_16x16x4_f32(
            /*neg_a=*/false, a, /*neg_b=*/false, bmat,
            /*c_mod=*/(short)0, acc, /*reuse_a=*/false, /*reuse_b=*/false);
    }
    const float bv = bias[col0 + lm];
#pragma unroll
    for (int i = 0; i < 8; ++i) {
        // C/D layout: VGPR i -> row i (lanes 0-15) / row i+8 (lanes 16-31), col = lane&15
        h[(size_t)(row0 + i + 8 * hi) * C_OUT + col0 + lm] = acc[i] + bv;
    }
}

// ---------------------------------------------------------------------------
// 4) BatchNorm stats (sum / sumsq per column) + fused normalize + exact GELU.
// ---------------------------------------------------------------------------
__global__ void zero_stats_kernel(float* __restrict__ stats) {
    stats[threadIdx.x] = 0.0f;   // 256 entries: sum[128], sumsq[128]
}

__global__ __launch_bounds__(256) void bn_stats_kernel(const float* __restrict__ h,
                                                       float* __restrict__ stats) {
    const int t = threadIdx.x;
    const int c = t & (C_OUT - 1);              // i = t + 256*j  =>  i & 127 == t & 127
    const size_t base = (size_t)blockIdx.x * 128 * C_OUT;   // 128 rows per block
    float s = 0.0f, q = 0.0f;
    for (int i = t; i < 128 * C_OUT; i += 256) {
        const float v = h[base + i];
        s += v; q += v * v;
    }
    __shared__ float ls[256], lq[256];
    ls[t] = s; lq[t] = q;
    __syncthreads();
    if (t < 128) {
        s = ls[t] + ls[t + 128];
        q = lq[t] + lq[t + 128];
        atomicAdd(&stats[c], s);
        atomicAdd(&stats[C_OUT + c], q);
    }
}

__global__ __launch_bounds__(256) void bn_gelu_kernel(float* __restrict__ h,
                                                      const float* __restrict__ stats,
                                                      const float* __restrict__ gamma,
                                                      const float* __restrict__ beta) {
    const size_t idx = (size_t)blockIdx.x * 256 + threadIdx.x;
    const int c = (int)(idx & (C_OUT - 1));
    const float inv = 1.0f / (float)(BB * NN);
    const float mu  = stats[c] * inv;
    const float var = stats[C_OUT + c] * inv - mu * mu;
    float v = h[idx];
    v = (v - mu) * rsqrtf(var + EPSBN) * gamma[c] + beta[c];
    v = 0.5f * v * (1.0f + erff(v * 0.70710678118654752f));   // exact GELU
    h[idx] = v;
}

// ---------------------------------------------------------------------------
// 5) Gather neighbor features, max over K; emit new_pos / new_batch.
// ---------------------------------------------------------------------------
__global__ __launch_bounds__(128) void gather_max_kernel(const float* __restrict__ h,
                                                         const float* __restrict__ pos,
                                                         const int* __restrict__ fps_idx,
                                                         const int* __restrict__ nbr,
                                                         float* __restrict__ new_x,
                                                         float* __restrict__ new_pos,
                                                         float* __restrict__ new_batch) {
    const int i = blockIdx.x;            // 0..B*M-1
    const int b = i / MM;
    const int t = threadIdx.x;           // channel

    __shared__ int snb[KK];
    __shared__ int sfi;
    if (t < KK) snb[t] = nbr[(size_t)i * KK + t];
    if (t == KK) sfi = fps_idx[i];
    __syncthreads();

    const float* hb = h + (size_t)b * NN * C_OUT;
    float m = -3.4e38f;
#pragma unroll
    for (int k = 0; k < KK; ++k) m = fmaxf(m, hb[(size_t)snb[k] * C_OUT + t]);
    new_x[(size_t)i * C_OUT + t] = m;

    if (t < 3)  new_pos[i * 3 + t] = pos[((size_t)b * NN + sfi) * 3 + t];
    if (t == 3) new_batch[i] = (float)b;
}

// ---------------------------------------------------------------------------
extern "C" void kernel_launch(void* const* d_in, const int* in_sizes, int n_in,
                              void* d_out, int out_size, void* d_ws, size_t ws_size,
                              hipStream_t stream) {
    const float* x     = (const float*)d_in[0];   // [B*N, C_IN]
    const float* pos   = (const float*)d_in[1];   // [B*N, 3]
    // d_in[2] = batch (unused; implied by layout)
    const float* W     = (const float*)d_in[3];   // [C_IN, C_OUT]
    const float* bias  = (const float*)d_in[4];   // [C_OUT]
    const float* gamma = (const float*)d_in[5];   // [C_OUT]
    const float* beta  = (const float*)d_in[6];   // [C_OUT]

    // workspace layout
    float* h      = (float*)d_ws;                                   // B*N*C_OUT f32
    int*   fpsidx = (int*)(h + (size_t)BB * NN * C_OUT);            // B*M int
    int*   nbr    = fpsidx + BB * MM;                               // B*M*K int
    float* stats  = (float*)(nbr + (size_t)BB * MM * KK);           // 2*C_OUT f32

    // output layout: new_x | new_pos | new_batch (concatenated flat)
    float* new_x     = (float*)d_out;
    float* new_pos   = new_x + (size_t)BB * MM * C_OUT;
    float* new_batch = new_pos + (size_t)BB * MM * 3;

    fps_kernel<<<BB, 1024, 0, stream>>>(pos, fpsidx);
    knn_kernel<<<dim3(MM / 256, BB), 256, 0, stream>>>(pos, fpsidx, nbr);

    const int n_tiles = (BB * NN / 16) * (C_OUT / 16);
    gemm_kernel<<<n_tiles, 32, 0, stream>>>(x, W, bias, h);

    zero_stats_kernel<<<1, 2 * C_OUT, 0, stream>>>(stats);
    bn_stats_kernel<<<(BB * NN) / 128, 256, 0, stream>>>(h, stats);
    bn_gelu_kernel<<<(BB * NN * C_OUT) / 256, 256, 0, stream>>>(h, stats, gamma, beta);

    gather_max_kernel<<<BB * MM, 128, 0, stream>>>(h, pos, fpsidx, nbr,
                                                   new_x, new_pos, new_batch);
}